// Scattering_v1_86560770883605
// MI455X (gfx1250) — compile-verified
//
#include <hip/hip_runtime.h>
#include <cstdint>

#define EPS 1e-07f

typedef float v2f __attribute__((ext_vector_type(2)));
typedef float v8f __attribute__((ext_vector_type(8)));

constexpr int   Bn = 524288;
constexpr int   Cn = 42;
constexpr int   Kn = 8;
constexpr size_t BC = (size_t)Bn * Cn;

// ---------------- LDS weight layout (floats) ----------------
// direct net: W1[C,5,9] b1[C,5] W2[C,4,5] b2[C,4] W3[C,4,4] b3[C,4] W4[C,3,4] b4[C,3]
// diffuse  : W1[C,5,8] ... (same tail)
#define O_dW1 0
#define O_db1 1890
#define O_dW2 2100
#define O_db2 2940
#define O_dW3 3108
#define O_db3 3780
#define O_dW4 3948
#define O_db4 4452
#define O_fW1 4578
#define O_fb1 6258
#define O_fW2 6468
#define O_fb2 7308
#define O_fW3 7476
#define O_fb3 8148
#define O_fW4 8316
#define O_fb4 8820
#define SMEM_FLOATS 8946

struct NetPtrs { const float* p[16]; };

// ---------------- helpers ----------------
__device__ __forceinline__ v8f wmma4(v2f a, v2f b, v8f c) {
    // D[16x16] = A[16x4] * B[4x16] + C   (f32, wave32)
    return __builtin_amdgcn_wmma_f32_16x16x4_f32(
        /*neg_a=*/false, a, /*neg_b=*/false, b,
        /*c_mod=*/(short)0, c, /*reuse_a=*/false, /*reuse_b=*/false);
}

// broadcast value held at lane (lane&15) to both wave halves
__device__ __forceinline__ float bcast16(float v, int ridx4) {
    int i = __builtin_amdgcn_ds_bpermute(ridx4, __float_as_int(v));
    return __int_as_float(i);
}

// unconditional A fragment: 2 consecutive LDS floats; padding garbage is
// killed by zeros on the B side (k >= IN) or by never reading D rows m >= OUT.
__device__ __forceinline__ v2f ldA(const float* p) {
    v2f a; a.x = p[0]; a.y = p[1]; return a;
}

// Layers 2..4 (5->4->4->3) + softmax + store. acc = layer-1 WMMA accumulation (pre-bias).
__device__ __forceinline__ void mlp_tail_store(
    const float* sm, int c, int ridx4, int hi, bool lo16,
    int l5, int l4,                     // lane offsets: r*5+2hi, r*4+2hi
    v8f acc,
    int oB1, int oW2, int oB2, int oW3, int oB3, int oW4, int oB4,
    float* outp)
{
    v8f zero = {};
    // layer1 bias + relu (valid at lanes 0..15; other lanes feed nothing)
    const float* b1p = sm + oB1 + c * 5;
    float h0 = fmaxf(acc[0] + b1p[0], 0.f);
    float h1 = fmaxf(acc[1] + b1p[1], 0.f);
    float h2 = fmaxf(acc[2] + b1p[2], 0.f);
    float h3 = fmaxf(acc[3] + b1p[3], 0.f);
    float h4 = fmaxf(acc[4] + b1p[4], 0.f);
    float H0 = bcast16(h0, ridx4), H1 = bcast16(h1, ridx4), H2 = bcast16(h2, ridx4),
          H3 = bcast16(h3, ridx4), H4 = bcast16(h4, ridx4);

    // layer2: 4 <- 5 (two K-chunks)
    const float* w2p = sm + oW2 + c * 20 + l5;
    v2f b20; b20.x = hi ? H2 : H0;  b20.y = hi ? H3 : H1;
    v2f b21; b21.x = hi ? 0.f : H4; b21.y = 0.f;
    v8f acc2 = wmma4(ldA(w2p),     b20, zero);
    acc2     = wmma4(ldA(w2p + 4), b21, acc2);
    const float* b2p = sm + oB2 + c * 4;
    float g0 = fmaxf(acc2[0] + b2p[0], 0.f);
    float g1 = fmaxf(acc2[1] + b2p[1], 0.f);
    float g2 = fmaxf(acc2[2] + b2p[2], 0.f);
    float g3 = fmaxf(acc2[3] + b2p[3], 0.f);
    float G0 = bcast16(g0, ridx4), G1 = bcast16(g1, ridx4),
          G2 = bcast16(g2, ridx4), G3 = bcast16(g3, ridx4);

    // layer3: 4 <- 4
    v2f b3v; b3v.x = hi ? G2 : G0; b3v.y = hi ? G3 : G1;
    v8f acc3 = wmma4(ldA(sm + oW3 + c * 16 + l4), b3v, zero);
    const float* b3p = sm + oB3 + c * 4;
    float q0 = fmaxf(acc3[0] + b3p[0], 0.f);
    float q1 = fmaxf(acc3[1] + b3p[1], 0.f);
    float q2 = fmaxf(acc3[2] + b3p[2], 0.f);
    float q3 = fmaxf(acc3[3] + b3p[3], 0.f);
    float Q0 = bcast16(q0, ridx4), Q1 = bcast16(q1, ridx4),
          Q2 = bcast16(q2, ridx4), Q3 = bcast16(q3, ridx4);

    // layer4: 3 <- 4
    v2f b4v; b4v.x = hi ? Q2 : Q0; b4v.y = hi ? Q3 : Q1;
    v8f acc4 = wmma4(ldA(sm + oW4 + c * 12 + l4), b4v, zero);
    const float* b4p = sm + oB4 + c * 3;
    float o0 = acc4[0] + b4p[0];
    float o1 = acc4[1] + b4p[1];
    float o2 = acc4[2] + b4p[2];

    if (lo16) {  // divergent store only; all WMMAs above ran with full EXEC
        float mx = fmaxf(fmaxf(o0, o1), o2);
        float e0 = expf(o0 - mx), e1 = expf(o1 - mx), e2 = expf(o2 - mx);
        float inv = 1.f / (e0 + e1 + e2);
        outp[0] = e0 * inv; outp[1] = e1 * inv; outp[2] = e2 * inv;
    }
}

// ---------------- kernel 1: streaming tau-sum + transmittances ----------------
__global__ __launch_bounds__(256) void trans_kernel(
    const float* __restrict__ tau, const float* __restrict__ mu_d,
    const float* __restrict__ mu_f, float* __restrict__ out)
{
    uint32_t idx = blockIdx.x * 256u + threadIdx.x;       // one (b,c) per thread
    uint32_t b = idx / (uint32_t)Cn;                      // magic-number div
    const float4* tp = reinterpret_cast<const float4*>(tau + (size_t)idx * Kn);
    float4 t0 = tp[0], t1 = tp[1];
    float s = t0.x + t0.y + t0.z + t0.w + t1.x + t1.y + t1.z + t1.w;
    out[idx]      = expf(-s / (mu_d[b] + EPS));
    out[BC + idx] = expf(-s / (mu_f[b] + EPS));
}

// ---------------- kernel 2: 2x42 tiny MLPs via f32 WMMA 16x16x4 ----------------
__global__ __launch_bounds__(256) void mlp_kernel(
    const float* __restrict__ mu_d, const float* __restrict__ mu_f,
    const float* __restrict__ cons, NetPtrs P,
    float* __restrict__ outD, float* __restrict__ outF)
{
    __shared__ float sm[SMEM_FLOATS];
    {
        const int segSize[16] = {1890,210,840,168,672,168,504,126,
                                 1680,210,840,168,672,168,504,126};
        const int segOff[16]  = {O_dW1,O_db1,O_dW2,O_db2,O_dW3,O_db3,O_dW4,O_db4,
                                 O_fW1,O_fb1,O_fW2,O_fb2,O_fW3,O_fb3,O_fW4,O_fb4};
        #pragma unroll
        for (int s = 0; s < 16; ++s)
            for (int i = threadIdx.x; i < segSize[s]; i += blockDim.x)
                sm[segOff[s] + i] = P.p[s][i];
    }
    __syncthreads();

    int lane = threadIdx.x & 31;
    int r    = lane & 15;          // batch row within tile / M row / N col
    int hi   = lane >> 4;          // which K pair this half-wave holds
    bool lo16 = lane < 16;
    int ridx4 = r << 2;            // ds_bpermute byte index for lane r
    int wave = (blockIdx.x * blockDim.x + threadIdx.x) >> 5;
    int b0   = wave * 16;
    int row  = b0 + r;

    // channel-independent lane offsets into the A-fragment weight rows
    int l9 = r * 9 + 2 * hi;       // direct W1  (stride 9)
    int l8 = r * 8 + 2 * hi;       // diffuse W1 (stride 8)
    int l5 = r * 5 + 2 * hi;       // W2 (stride 5)
    int l4 = r * 4 + 2 * hi;       // W3/W4 (stride 4)

    float mud = mu_d[row];
    float inv_md = 1.f / (mud + EPS);

    const float* crow = cons + (size_t)row * Kn;
    float c_a = crow[2 * hi + 0];
    float c_b = crow[2 * hi + 1];
    float c_c = crow[2 * hi + 4];
    float c_d = crow[2 * hi + 5];

    // B fragments (features x 16 batch cols), channel-independent, built once
    v2f Bd0; Bd0.x = c_a * inv_md;      Bd0.y = c_b * inv_md;   // feats 0..3
    v2f Bd1; Bd1.x = c_c * inv_md;      Bd1.y = c_d * inv_md;   // feats 4..7
    v2f Bd2; Bd2.x = hi ? 0.f : mud;    Bd2.y = 0.f;            // feat 8 = mu_direct
    v2f Bf0; Bf0.x = c_a;               Bf0.y = c_b;
    v2f Bf1; Bf1.x = c_c;               Bf1.y = c_d;

    v8f zero = {};
    #pragma unroll 2
    for (int c = 0; c < Cn; ++c) {
        // ---- direct net, layer1: 5 <- 9 (three K-chunks) ----
        const float* wd1 = sm + O_dW1 + c * 45 + l9;
        v8f accD = wmma4(ldA(wd1),     Bd0, zero);
        accD     = wmma4(ldA(wd1 + 4), Bd1, accD);
        accD     = wmma4(ldA(wd1 + 8), Bd2, accD);
        mlp_tail_store(sm, c, ridx4, hi, lo16, l5, l4, accD,
                       O_db1, O_dW2, O_db2, O_dW3, O_db3, O_dW4, O_db4,
                       outD + ((size_t)row * Cn + c) * 3);

        // ---- diffuse net, layer1: 5 <- 8 (two K-chunks) ----
        const float* wf1 = sm + O_fW1 + c * 40 + l8;
        v8f accF = wmma4(ldA(wf1),     Bf0, zero);
        accF     = wmma4(ldA(wf1 + 4), Bf1, accF);
        mlp_tail_store(sm, c, ridx4, hi, lo16, l5, l4, accF,
                       O_fb1, O_fW2, O_fb2, O_fW3, O_fb3, O_fW4, O_fb4,
                       outF + ((size_t)row * Cn + c) * 3);
    }
}

// ---------------- launch ----------------
extern "C" void kernel_launch(void* const* d_in, const int* in_sizes, int n_in,
                              void* d_out, int out_size, void* d_ws, size_t ws_size,
                              hipStream_t stream) {
    (void)in_sizes; (void)n_in; (void)d_ws; (void)ws_size; (void)out_size;
    const float* tau  = (const float*)d_in[0];
    const float* mu_d = (const float*)d_in[1];
    const float* mu_f = (const float*)d_in[2];
    const float* cons = (const float*)d_in[3];
    NetPtrs P;
    for (int i = 0; i < 16; ++i) P.p[i] = (const float*)d_in[4 + i];
    float* out = (float*)d_out;

    // out layout: [t_direct | t_diffuse | e_split_direct | e_split_diffuse]
    float* outD = out + 2 * BC;   // e_split_direct  [B,C,3]
    float* outF = out + 5 * BC;   // e_split_diffuse [B,C,3]

    // 1) streaming transmittance: B*C threads, fully coalesced 32B/row reads
    trans_kernel<<<(uint32_t)(BC / 256), 256, 0, stream>>>(tau, mu_d, mu_f, out);

    // 2) WMMA MLPs: one wave32 per 16-row tile, 13 f32-WMMAs per channel
    uint32_t waves  = Bn / 16;                 // 32768
    uint32_t blocks = waves * 32 / 256;        // 4096
    mlp_kernel<<<blocks, 256, 0, stream>>>(mu_d, mu_f, cons, P, outD, outF);
}